// DSSM_24275155157030
// MI455X (gfx1250) — compile-verified
//
#include <hip/hip_runtime.h>

// ---------------------------------------------------------------------------
// DSS / S4D forward on MI455X (gfx1250) — fused chunked-scan, one workgroup
// per (b,h).
//
//   y[b,h,t] = mask(t<len[b]) * ( sum_{s<=t} K[h,s] u[b,h,t-s] + D[h] u[b,h,t] )
//
// Roofline: mandatory HBM traffic = u(67MB) + y(67MB); per-head operand
// matrices (~20MB) are L2-resident (192MB L2) and reused by all 16 batches.
// The fused workgroup keeps the S (chunk summaries) and X (chunk states)
// intermediates in LDS instead of 540MB of HBM round-trips: ~155MB @ 23.3TB/s
// ~= 6.6us. Compute (~1e10 FLOPs) is negligible => full-precision f32 WMMA.
//
// This revision:
//  * operand matrices stored TRANSPOSED so each WMMA B-fragment is one
//    contiguous global_load_b64 (was 2 strided b32)
//  * one column-tile per wave with B fragments cached in registers across
//    row tiles (B reuse in VGPRs, no redundant L2 reads)
//  * LDS row strides padded (64->68, 128->132 floats) so the 16 A-fragment
//    rows map to 16 distinct banks (stride 4 mod 64) instead of 1.
//
// CDNA5 paths: v_wmma_f32_16x16x4_f32 (wave32), global_load_async_to_lds_b128
// (ASYNCcnt) for staging u, ds_load_b64 fragments, ds_add_f32 in precompute.
// ---------------------------------------------------------------------------

typedef float v2f __attribute__((ext_vector_type(2)));
typedef float v8f __attribute__((ext_vector_type(8)));

#define Bsz 16
#define Hsz 256
#define Nsz 64
#define Lsz 4096
#define Qc  64              // chunk length
#define Mc  64              // number of chunks = L/Q
#define J2  (2*Nsz)         // 128 = [Re | Im] state width

// Workspace layout (floats) — transposed per-head operand matrices.
#define OFF_BS  0                              // H*J2*Qc : BsT[h][n2][s]
#define OFF_E   (OFF_BS + Hsz*J2*Qc)           // H*Qc*J2 : ET[h][r][j]
#define OFF_T   (OFF_E  + Hsz*Qc*J2)           // H*Qc*Qc : TT[h][r][s]
#define OFF_WC  (OFF_T  + Hsz*Qc*Qc)           // H*N*2   : w^Q per (h,n)

// LDS layout (floats), bank-conflict-padded row strides
#define UP  68                          // u rows: 64 cols + 4 pad
#define SP  132                         // S/X rows: 128 cols + 4 pad
#define LDS_U   0                       // 64*68  = 4352
#define LDS_S   (LDS_U + Mc*UP)         // 64*132 = 8448
#define LDS_X   (LDS_S + Mc*SP)         // 64*132 = 8448
#define LDS_FLOATS (LDS_X + Mc*SP)      // 21248 floats = 83KB

// ---------------------------------------------------------------------------
// Phase 0: per-(h,n) parameter cook. Builds transposed GEMM operand matrices.
// grid = H blocks, 64 threads (thread == n).
// ---------------------------------------------------------------------------
__global__ void dssm_precompute(const float* __restrict__ log_dt,
                                const float* __restrict__ C,
                                const float* __restrict__ log_A_real,
                                const float* __restrict__ A_imag,
                                const float* __restrict__ Dvec,
                                float* __restrict__ ws) {
  const int h = blockIdx.x;
  const int n = threadIdx.x;           // 0..63
  __shared__ float Ksh[Qc];
  Ksh[n] = 0.f;
  __syncthreads();

  const float dt = __expf(log_dt[h]);
  const float Ar = -__expf(log_A_real[h*Nsz + n]);
  const float Ai = A_imag[h*Nsz + n];
  const float ar = Ar*dt, ai = Ai*dt;          // dtA (complex)
  const float ea = __expf(ar);
  const float wr = ea*__cosf(ai), wi = ea*__sinf(ai);   // w = exp(dtA)
  // q = (w - 1)/A
  const float nr = wr - 1.f, ni = wi;
  const float inv = 1.f/(Ar*Ar + Ai*Ai);
  const float qr = (nr*Ar + ni*Ai)*inv;
  const float qi = (ni*Ar - nr*Ai)*inv;
  // Ct = Cc * q
  const float cr = C[(h*Nsz + n)*2 + 0], ci = C[(h*Nsz + n)*2 + 1];
  const float ctr = cr*qr - ci*qi;
  const float cti = cr*qi + ci*qr;

  // w^Q for the inter-chunk scan
  {
    const float e = __expf(ar*(float)Qc);
    ws[OFF_WC + (h*Nsz + n)*2 + 0] = e*__cosf(ai*(float)Qc);
    ws[OFF_WC + (h*Nsz + n)*2 + 1] = e*__sinf(ai*(float)Qc);
  }

  // BsT[h][n][s] = Re(w^(Q-1-s)),  BsT[h][64+n][s] = Im(w^(Q-1-s))
  for (int s = 0; s < Qc; ++s) {
    const float p = (float)(Qc - 1 - s);
    const float e = __expf(ar*p);
    ws[OFF_BS + (h*J2 + n)*Qc + s]       = e*__cosf(ai*p);
    ws[OFF_BS + (h*J2 + Nsz + n)*Qc + s] = e*__sinf(ai*p);
  }

  // ET[h][r][n] = 2 Re(Ct w^(r+1)),  ET[h][r][64+n] = -2 Im(Ct w^(r+1)).
  // Also reduce K[d] = sum_n 2 Re(Ct w^d) into LDS (ds_add_f32 atomics).
  for (int r = 0; r < Qc; ++r) {
    const float p = (float)(r + 1);
    const float e = __expf(ar*p);
    const float pr = e*__cosf(ai*p), pi = e*__sinf(ai*p);
    ws[OFF_E + (h*Qc + r)*J2 + n]       =  2.f*(ctr*pr - cti*pi);
    ws[OFF_E + (h*Qc + r)*J2 + Nsz + n] = -2.f*(ctr*pi + cti*pr);
    const float pd = (float)r;
    const float ed = __expf(ar*pd);
    const float kdr = ctr*(ed*__cosf(ai*pd)) - cti*(ed*__sinf(ai*pd));
    atomicAdd(&Ksh[r], 2.f*kdr);
  }
  __syncthreads();

  // Toeplitz operand, transposed: TT[h][r][s] = K[r-s] for r>=s, + D on diag.
  // thread n plays the role of s.
  const int s = n;
  const float Dh = Dvec[h];
  for (int r = 0; r < Qc; ++r) {
    float v = (r >= s) ? Ksh[r - s] : 0.f;
    if (r == s) v += Dh;
    ws[OFF_T + (h*Qc + r)*Qc + s] = v;
  }
}

// ---------------------------------------------------------------------------
// Fused kernel: one workgroup (256 thr = 8 waves) per (b,h).
//   A) async-stage u[b,h,:] (16KB) into padded LDS (global_load_async_to_lds)
//   B) S(64x128) = U(64x64) x Bs(64x128)           (WMMA, wave = col tile)
//   C) X[c+1] = w^Q X[c] + S[c] scan in LDS        (64 threads x 64 steps)
//   D) Y(64x64) = U x T + X x E, mask, store       (WMMA, B cached in VGPRs)
//
// V_WMMA_F32_16X16X4_F32 fragment layout (wave32):
//   A[16x4]: a.x=A[lane&15][k0], a.y=A[lane&15][k0+1], k0=(lane>>4)*2
//   B[4x16]: b.x=B[k0][lane&15], b.y=B[k0+1][lane&15]
//   C/D[16x16]: c[i]=C[i+(lane>>4)*8][lane&15]
// ---------------------------------------------------------------------------
__global__ void __launch_bounds__(256)
dssm_fused(const float* __restrict__ u,
           const float* __restrict__ ws,
           const int* __restrict__ length,
           float* __restrict__ out) {
  __shared__ __align__(16) float smem[LDS_FLOATS];

  const int tid  = threadIdx.x;
  const int lane = tid & 31;
  const int wave = tid >> 5;
  const int l15  = lane & 15;
  const int half = lane >> 4;            // 0/1
  const int khalf = half << 1;           // K sub-offset 0/2
  const int h  = blockIdx.x & (Hsz - 1);
  const int bb = blockIdx.x >> 8;

  const float* uBase   = u   + (size_t)blockIdx.x * Lsz;
  float*       outBase = out + (size_t)blockIdx.x * Lsz;

  // ---- Phase A: async copy of u slice into LDS, padded rows of UP floats --
  // LDS dest address is wave-relative; smem is the only LDS object -> base 0.
#pragma unroll
  for (int j = 0; j < 4; ++j) {
    const unsigned elem = (unsigned)tid*4u + (unsigned)j*1024u;  // float index
    const unsigned row  = elem >> 6, colu = elem & 63u;
    const unsigned ldsByte = (LDS_U + row*UP + colu)*4u;         // 16B aligned
    const float* gp = uBase + elem;
    asm volatile("global_load_async_to_lds_b128 %0, %1, off"
                 :: "v"(ldsByte), "v"(gp) : "memory");
  }
  asm volatile("s_wait_asynccnt 0x0" ::: "memory");
  __syncthreads();

  // ---- Phase B: S = U x Bs. wave = column tile (8 x 16 cols = 128).
  //      B fragments (16 x v2f) cached in registers across 4 row tiles. -----
  {
    const int col = (wave << 4) + l15;
    const float* bsRow = ws + OFF_BS + ((size_t)h*J2 + col)*Qc;  // BsT[h][col]
    v2f bf[16];
#pragma unroll
    for (int k4 = 0; k4 < 16; ++k4)
      bf[k4] = *(const v2f*)(bsRow + k4*4 + khalf);   // global_load_b64

#pragma unroll
    for (int rt = 0; rt < 4; ++rt) {
      const int c0row = rt << 4;
      const int arow  = LDS_U + (c0row + l15)*UP;
      v8f acc = {};
#pragma unroll
      for (int k4 = 0; k4 < 16; ++k4) {
        v2f a = *(const v2f*)(&smem[arow + k4*4 + khalf]);   // ds_load_b64
        acc = __builtin_amdgcn_wmma_f32_16x16x4_f32(false, a, false, bf[k4],
                                                    (short)0, acc, false, false);
      }
#pragma unroll
      for (int i = 0; i < 8; ++i) {
        const int r = c0row + i + (half << 3);
        smem[LDS_S + r*SP + col] = acc[i];
      }
    }
  }
  __syncthreads();

  // ---- Phase C: inter-chunk scan in LDS (thread n = state index) ---------
  if (tid < Nsz) {
    const int n = tid;
    const float wcr = ws[OFF_WC + (h*Nsz + n)*2 + 0];
    const float wci = ws[OFF_WC + (h*Nsz + n)*2 + 1];
    float xr = 0.f, xi = 0.f;
    for (int c = 0; c < Mc; ++c) {
      smem[LDS_X + c*SP + n]       = xr;
      smem[LDS_X + c*SP + Nsz + n] = xi;
      const float sr = smem[LDS_S + c*SP + n];
      const float si = smem[LDS_S + c*SP + Nsz + n];
      const float nxr = wcr*xr - wci*xi + sr;
      const float nxi = wcr*xi + wci*xr + si;
      xr = nxr; xi = nxi;
    }
  }
  __syncthreads();

  // ---- Phase D: Y = U x T + X x E. wave -> (col tile, row-tile pair);
  //      48 B fragments cached in registers across 2 row tiles. -------------
  {
    const int lenb    = length[bb];
    const int colTile = wave & 3;
    const int rbase   = (wave >> 2) << 1;      // row tiles rbase, rbase+1
    const int col     = (colTile << 4) + l15;
    const float* tRow = ws + OFF_T + ((size_t)h*Qc + col)*Qc;  // TT[h][col]
    const float* eRow = ws + OFF_E + ((size_t)h*Qc + col)*J2;  // ET[h][col]

    v2f bt[16];
#pragma unroll
    for (int k4 = 0; k4 < 16; ++k4)
      bt[k4] = *(const v2f*)(tRow + k4*4 + khalf);
    v2f be[32];
#pragma unroll
    for (int k4 = 0; k4 < 32; ++k4)
      be[k4] = *(const v2f*)(eRow + k4*4 + khalf);

#pragma unroll
    for (int rt = rbase; rt < rbase + 2; ++rt) {
      const int c0row = rt << 4;
      const int arowU = LDS_U + (c0row + l15)*UP;
      const int arowX = LDS_X + (c0row + l15)*SP;
      v8f acc = {};
#pragma unroll
      for (int k4 = 0; k4 < 16; ++k4) {        // intra-chunk Toeplitz
        v2f a = *(const v2f*)(&smem[arowU + k4*4 + khalf]);
        acc = __builtin_amdgcn_wmma_f32_16x16x4_f32(false, a, false, bt[k4],
                                                    (short)0, acc, false, false);
      }
#pragma unroll
      for (int k4 = 0; k4 < 32; ++k4) {        // inter-chunk state readout
        v2f a = *(const v2f*)(&smem[arowX + k4*4 + khalf]);
        acc = __builtin_amdgcn_wmma_f32_16x16x4_f32(false, a, false, be[k4],
                                                    (short)0, acc, false, false);
      }
#pragma unroll
      for (int i = 0; i < 8; ++i) {
        const int c = c0row + i + (half << 3);
        const int t = c*Qc + col;
        outBase[t] = (t < lenb) ? acc[i] : 0.f;
      }
    }
  }
}

// ---------------------------------------------------------------------------
extern "C" void kernel_launch(void* const* d_in, const int* in_sizes, int n_in,
                              void* d_out, int out_size, void* d_ws, size_t ws_size,
                              hipStream_t stream) {
  (void)in_sizes; (void)n_in; (void)out_size; (void)ws_size;
  const float* u          = (const float*)d_in[0];
  const float* log_dt     = (const float*)d_in[1];
  const float* C          = (const float*)d_in[2];
  const float* log_A_real = (const float*)d_in[3];
  const float* A_imag     = (const float*)d_in[4];
  const float* D          = (const float*)d_in[5];
  const int*   length     = (const int*)d_in[6];
  float* ws  = (float*)d_ws;
  float* out = (float*)d_out;

  dssm_precompute<<<Hsz, 64, 0, stream>>>(log_dt, C, log_A_real, A_imag, D, ws);
  dssm_fused<<<Bsz*Hsz, 256, 0, stream>>>(u, ws, length, out);
}